// auxiliary_y_fixed_9947144257678
// MI455X (gfx1250) — compile-verified
//
#include <hip/hip_runtime.h>
#include <hip/hip_bf16.h>

typedef float v2f __attribute__((ext_vector_type(2)));
typedef float v8f __attribute__((ext_vector_type(8)));
typedef float v4f __attribute__((ext_vector_type(4)));

// -------------------------------------------------------------------------
// Kernel 1: loc = z_ins @ W + b via V_WMMA_F32_16X16X4_F32 (exact fp32).
// One wave (32 lanes) per 16-row tile. B matrix holds W replicated in every
// column, so D[m, n] == loc[m] for all n. Per ISA layout, lane 0 holds
// D[0..7, 0] in acc[0..7] and lane 16 holds D[8..15, 0].
// -------------------------------------------------------------------------
__global__ void matvec_wmma_kernel(const float* __restrict__ z,
                                   const float* __restrict__ W,
                                   const float* __restrict__ bias,
                                   float* __restrict__ loc,
                                   int n_tiles, int D) {
    int wave = (int)((blockIdx.x * blockDim.x + threadIdx.x) >> 5);
    int lane = (int)(threadIdx.x & 31);
    if (wave >= n_tiles) return;          // wave-uniform branch; EXEC stays all-1s

    int m = lane & 15;                    // row within tile
    int h = lane >> 4;                    // lane-half selects K pair (0,1) vs (2,3)
    int base = wave * 16;

    const float* rowp = z + (size_t)(base + m) * (size_t)D + 2 * h;

    v8f acc = {0.f, 0.f, 0.f, 0.f, 0.f, 0.f, 0.f, 0.f};

    #pragma unroll 8
    for (int k = 0; k < D; k += 4) {
        // A fragment: z[base+m][k+2h], z[base+m][k+2h+1]  (8B aligned)
        v2f a = *(const v2f*)(rowp + k);
        // B fragment: W broadcast across all 16 columns
        v2f bfrag;
        bfrag.x = W[k + 2 * h];
        bfrag.y = W[k + 2 * h + 1];
        acc = __builtin_amdgcn_wmma_f32_16x16x4_f32(
                  /*neg_a=*/false, a,
                  /*neg_b=*/false, bfrag,
                  /*c_mod=*/(short)0, acc,
                  /*reuse_a=*/false, /*reuse_b=*/false);
    }

    float bb = bias[0];
    if (m == 0) {                         // lanes 0 and 16
        float* outp = loc + base + 8 * h; // lane0 -> rows 0..7, lane16 -> rows 8..15
        #pragma unroll
        for (int r = 0; r < 8; ++r) outp[r] = acc[r] + bb;
    }
}

// Scalar tail for N not divisible by 16 (defensive; N=250000 is divisible).
__global__ void matvec_tail_kernel(const float* __restrict__ z,
                                   const float* __restrict__ W,
                                   const float* __restrict__ bias,
                                   float* __restrict__ loc,
                                   int start, int N, int D) {
    int i = start + (int)(blockIdx.x * blockDim.x + threadIdx.x);
    if (i >= N) return;
    float s = bias[0];
    for (int k = 0; k < D; ++k) s += z[(size_t)i * D + k] * W[k];
    loc[i] = s;
}

// -------------------------------------------------------------------------
// Segment argmax machinery. Monotone float->uint key so u32 atomicMax works
// for negative floats too.
// -------------------------------------------------------------------------
__device__ __forceinline__ unsigned fkey(float f) {
    unsigned u = __float_as_uint(f);
    return (u & 0x80000000u) ? ~u : (u | 0x80000000u);
}

__global__ void init_ws_kernel(unsigned* __restrict__ segkey,
                               int* __restrict__ arg, int B, int N) {
    int i = (int)(blockIdx.x * blockDim.x + threadIdx.x);
    if (i < B) { segkey[i] = 0u; arg[i] = N; }
}

__global__ void seg_max_kernel(const float* __restrict__ loc,
                               const int* __restrict__ bag,
                               unsigned* __restrict__ segkey, int N) {
    int i = (int)(blockIdx.x * blockDim.x + threadIdx.x);
    if (i < N) atomicMax(&segkey[bag[i]], fkey(loc[i]));
}

__global__ void seg_arg_kernel(const float* __restrict__ loc,
                               const int* __restrict__ bag,
                               const unsigned* __restrict__ segkey,
                               int* __restrict__ arg, int N) {
    int i = (int)(blockIdx.x * blockDim.x + threadIdx.x);
    if (i < N) {
        int b = bag[i];
        if (fkey(loc[i]) == segkey[b]) atomicMin(&arg[b], i);
    }
}

// -------------------------------------------------------------------------
// Gather: one 128-thread block per bag. Empty bags write zeros.
// -------------------------------------------------------------------------
__global__ void gather_kernel(const int* __restrict__ arg,
                              const float* __restrict__ loc,
                              const float* __restrict__ z,
                              const float* __restrict__ bi,
                              const float* __restrict__ mu,
                              const float* __restrict__ sd,
                              float* __restrict__ oM,
                              float* __restrict__ oInst,
                              float* __restrict__ oZ,
                              float* __restrict__ oMu,
                              float* __restrict__ oSd,
                              int N, int D, int F) {
    int j = (int)blockIdx.x;
    int t = (int)threadIdx.x;            // 0..127
    int a = arg[j];
    bool empty = (a >= N);
    int ac = empty ? 0 : a;

    if (t == 0) oM[j] = empty ? 0.f : loc[ac];

    // D-wide rows (D == blockDim.x == 128)
    if (t < D) {
        oZ [(size_t)j * D + t] = empty ? 0.f : z [(size_t)ac * D + t];
        oMu[(size_t)j * D + t] = empty ? 0.f : mu[(size_t)ac * D + t];
        oSd[(size_t)j * D + t] = empty ? 0.f : sd[(size_t)ac * D + t];
    }

    // F-wide row via float4 (F == 512 == 128 threads * 4)
    int nv4 = F >> 2;
    for (int v = t; v < nv4; v += blockDim.x) {
        v4f val = {0.f, 0.f, 0.f, 0.f};
        if (!empty) val = ((const v4f*)(bi + (size_t)ac * F))[v];
        ((v4f*)(oInst + (size_t)j * F))[v] = val;
    }
}

// -------------------------------------------------------------------------
extern "C" void kernel_launch(void* const* d_in, const int* in_sizes, int n_in,
                              void* d_out, int out_size, void* d_ws, size_t ws_size,
                              hipStream_t stream) {
    const float* z_ins         = (const float*)d_in[0];
    const int*   bag_idx       = (const int*)  d_in[1];
    const float* bag_instances = (const float*)d_in[2];
    const float* instance_mu   = (const float*)d_in[3];
    const float* instance_std  = (const float*)d_in[4];
    const float* W             = (const float*)d_in[5];
    const float* bias          = (const float*)d_in[6];
    float*       out           = (float*)d_out;

    const int N = in_sizes[1];                 // 250000
    const int D = in_sizes[0] / N;             // 128
    const int F = in_sizes[2] / N;             // 512
    const int B = (out_size - N) / (1 + F + 3 * D);  // 5000

    // Output layout (flat, tuple return order)
    float* oM    = out;                                  // B
    float* oInst = oM    + B;                            // B*F
    float* oZ    = oInst + (size_t)B * F;                // B*D
    float* oLoc  = oZ    + (size_t)B * D;                // N   (loc_ins)
    float* oMu   = oLoc  + N;                            // B*D
    float* oSd   = oMu   + (size_t)B * D;                // B*D

    // Workspace
    unsigned* segkey = (unsigned*)d_ws;                  // B u32
    int*      arg    = (int*)(segkey + B);               // B i32
    (void)ws_size; (void)n_in;

    // 1) matvec -> loc_ins (WMMA path for full 16-row tiles)
    int tiles = N / 16;
    if (tiles > 0) {
        int blocks = (tiles + 7) / 8;        // 8 waves of 32 per 256-thread block
        matvec_wmma_kernel<<<blocks, 256, 0, stream>>>(z_ins, W, bias, oLoc, tiles, D);
    }
    int tail_start = tiles * 16;
    if (tail_start < N) {
        int tail = N - tail_start;
        matvec_tail_kernel<<<(tail + 255) / 256, 256, 0, stream>>>(
            z_ins, W, bias, oLoc, tail_start, N, D);
    }

    // 2) init segment state
    init_ws_kernel<<<(B + 255) / 256, 256, 0, stream>>>(segkey, arg, B, N);

    // 3) per-bag max (monotone-key u32 atomics)
    seg_max_kernel<<<(N + 255) / 256, 256, 0, stream>>>(oLoc, bag_idx, segkey, N);

    // 4) per-bag argmax with lowest-index tiebreak
    seg_arg_kernel<<<(N + 255) / 256, 256, 0, stream>>>(oLoc, bag_idx, segkey, arg, N);

    // 5) gather winning rows (zeros for empty bags)
    gather_kernel<<<B, 128, 0, stream>>>(arg, oLoc, z_ins, bag_instances,
                                         instance_mu, instance_std,
                                         oM, oInst, oZ, oMu, oSd, N, D, F);
}